// QKernel_64948495450123
// MI455X (gfx1250) — compile-verified
//
#include <hip/hip_runtime.h>
#include <math.h>

// Quantum circuit expectation kernel for MI455X (gfx1250, wave32, WMMA).
//
// Math restructuring:
//   state(1024) viewed as S[32][32]; all 10 Rot gates == Uhi(32x32) * S * Ulo^T(32x32)
//   (Uhi = R0x..xR4, Ulo = R5x..xR9 Kronecker products, complex).
//   CNOT ring folded into the reduction as a GF(2) index map (zero cost).
//   Both GEMMs run on V_WMMA_F32_16X16X4_F32 tiles; state rows staged into LDS
//   with GLOBAL_LOAD_ASYNC_TO_LDS_B128 (ASYNCcnt) when the toolchain exposes it.

typedef __attribute__((ext_vector_type(2))) float v2f;
typedef __attribute__((ext_vector_type(8))) float v8f;
typedef int v4i __attribute__((vector_size(16)));
typedef __attribute__((address_space(1))) v4i gv4i;   // global-AS 16B vector
typedef __attribute__((address_space(3))) v4i lv4i;   // LDS-AS 16B vector

#define LS  36   // padded LDS row stride (words). 36*4B rows are 16B-aligned for
                 // async B128 LDS writes; 36n mod 64 is injective for n=0..15 so
                 // transposed fragment reads stay bank-conflict-free.
#define WPB 2    // waves per block

#if __has_builtin(__builtin_amdgcn_global_load_async_to_lds_b128)
#define USE_ASYNC_LDS 1
#endif

__device__ __forceinline__ v8f wmma_f32(v2f a, v2f b, v8f c) {
  // D = A(16x4) * B(4x16) + C(16x16), fp32. No NEG modifiers (F32 NEG is C-neg only).
  return __builtin_amdgcn_wmma_f32_16x16x4_f32(false, a, false, b, (short)0, c,
                                               false, false);
}

// A-fragment: 16x4 fp32. lanes 0-15 -> M=0..15 (K=0,1 in the 2 VGPRs),
// lanes 16-31 -> same M, K=2,3.
__device__ __forceinline__ v2f ldA(const float* G, int tm, int ks, int lm, int half) {
  const float* p = G + (tm * 16 + lm) * LS + ks * 4 + 2 * half;
  v2f r; r.x = p[0]; r.y = p[1];
  return r;
}

// B-fragment: 4x16 fp32. lanes 0-15 -> N=0..15 (K=0,1), lanes 16-31 -> K=2,3.
template <bool TRANS>
__device__ __forceinline__ v2f ldB(const float* X, int ks, int tn, int lm, int half) {
  const int k0 = ks * 4 + 2 * half;
  const int n  = tn * 16 + lm;
  v2f r;
  if (TRANS) {           // B[k][n] = X[n][k]
    const float* p = X + n * LS + k0;
    r.x = p[0]; r.y = p[1];
  } else {               // B[k][n] = X[k][n]
    r.x = X[k0 * LS + n];
    r.y = X[(k0 + 1) * LS + n];
  }
  return r;
}

// C/D tile: VGPR i -> M = i (+8 for upper lane half), N = lane%16.
__device__ __forceinline__ void stD(float* D, int tm, int tn, int lm, int half, v8f acc) {
  const int N = tn * 16 + lm;
#pragma unroll
  for (int i = 0; i < 8; ++i) {
    const int M = tm * 16 + i + 8 * half;
    D[M * LS + N] = acc[i];
  }
}

// D = (Ar + i*Ai) * (Br + i*Bi), all 32x32, via 16x16 WMMA tiles.
// BIMAG=false: B is real (Bi ignored).  TRANS: use B^T as the right operand.
template <bool BIMAG, bool TRANS>
__device__ __forceinline__ void cgemm(const float* Ar, const float* Ai,
                                      const float* Br, const float* Bi,
                                      float* Dr, float* Di, int lm, int half) {
#pragma unroll
  for (int tm = 0; tm < 2; ++tm) {
#pragma unroll
    for (int tn = 0; tn < 2; ++tn) {
      v8f accR = {0.f, 0.f, 0.f, 0.f, 0.f, 0.f, 0.f, 0.f};
      v8f accI = {0.f, 0.f, 0.f, 0.f, 0.f, 0.f, 0.f, 0.f};
#pragma unroll
      for (int ks = 0; ks < 8; ++ks) {
        v2f ar = ldA(Ar, tm, ks, lm, half);
        v2f ai = ldA(Ai, tm, ks, lm, half);
        v2f br = ldB<TRANS>(Br, ks, tn, lm, half);
        accR = wmma_f32(ar, br, accR);       // Re += ArBr
        accI = wmma_f32(ai, br, accI);       // Im += AiBr
        if (BIMAG) {
          v2f bi = ldB<TRANS>(Bi, ks, tn, lm, half);
          v2f nai; nai.x = -ai.x; nai.y = -ai.y;
          accR = wmma_f32(nai, bi, accR);    // Re -= AiBi
          accI = wmma_f32(ar,  bi, accI);    // Im += ArBi
        }
      }
      stD(Dr, tm, tn, lm, half, accR);
      stD(Di, tm, tn, lm, half, accI);
    }
  }
}

__device__ __forceinline__ float fixv(float x) {
  if (isnan(x)) return 0.f;
  if (isinf(x)) return x > 0.f ? 1.f : -1.f;
  return x;
}

__global__ __launch_bounds__(WPB * 32) void qcirc_kernel(
    const float* __restrict__ patch, const float* __restrict__ wts,
    float* __restrict__ out, int B) {
  __shared__ float sUhiR[32 * LS], sUhiI[32 * LS];
  __shared__ float sUloR[32 * LS], sUloI[32 * LS];
  __shared__ float sXr[WPB][32 * LS], sXi[WPB][32 * LS];
  __shared__ float sYr[WPB][32 * LS], sYi[WPB][32 * LS];

  const int lane = threadIdx.x & 31;
  const int wv   = threadIdx.x >> 5;
  const int half = lane >> 4;
  const int lm   = lane & 15;

  const int b  = blockIdx.x * WPB + wv;
  const int be = b < B ? b : (B > 0 ? B - 1 : 0);
  const float* row = patch + (size_t)be * 1024;

  float* Xr = sXr[wv]; float* Xi = sXi[wv];
  float* Yr = sYr[wv]; float* Yi = sYi[wv];

  // ---- Phase 0: kick off this wave's 4KB state-row fetch so the HBM access
  // overlaps the Kronecker build below.
#ifdef USE_ASYNC_LDS
  // CDNA5 async DMA: global -> LDS, tracked by ASYNCcnt, no VGPR data movement.
#pragma unroll
  for (int chunk = 0; chunk < 8; ++chunk) {
    const int j = chunk * 128 + lane * 4;   // 16B per lane per chunk
    gv4i* g = (gv4i*)(row + j);
    lv4i* l = (lv4i*)(Xr + (j >> 5) * LS + (j & 31));
    __builtin_amdgcn_global_load_async_to_lds_b128(g, l, 0, 0);
  }
#else
  __builtin_prefetch(row + lane * 32, 0, 0);
#endif

  // ---- Phase 1: build Uhi = R0x..xR4 and Ulo = R5x..xR9 in LDS ----
  // Rot(phi,theta,omega) = [[e^{-i(p+o)/2} c, -e^{i(p-o)/2} s],
  //                         [e^{-i(p-o)/2} s,  e^{i(p+o)/2} c]]
  // Weights are tiny rotations; hardware v_sin/v_cos (via __sinf/__cosf) keeps
  // the build VALU-only instead of pulling in ocml's scalar-heavy libm paths.
  float Rr[10][4], Ri[10][4];
#pragma unroll
  for (int w = 0; w < 10; ++w) {
    const float phi = wts[w * 3 + 0], th = wts[w * 3 + 1], om = wts[w * 3 + 2];
    const float ct = __cosf(0.5f * th), st = __sinf(0.5f * th);
    const float aa = 0.5f * (phi + om), bb = 0.5f * (phi - om);
    const float ca = __cosf(aa), sa = __sinf(aa);
    const float cb = __cosf(bb), sb = __sinf(bb);
    Rr[w][0] =  ct * ca; Ri[w][0] = -ct * sa;   // R00 = ep*c
    Rr[w][1] = -st * cb; Ri[w][1] = -st * sb;   // R01 = -em*s
    Rr[w][2] =  st * cb; Ri[w][2] = -st * sb;   // R10 = conj(em)*s
    Rr[w][3] =  ct * ca; Ri[w][3] =  ct * sa;   // R11 = conj(ep)*c
  }
  for (int idx = threadIdx.x; idx < 1024; idx += WPB * 32) {
    const int r = idx >> 5, c = idx & 31;
    float pr = 1.f, pi = 0.f;
#pragma unroll
    for (int w = 0; w < 5; ++w) {          // wire w (0..4) <-> bit (4-w) of r/c
      const int i = (r >> (4 - w)) & 1, j = (c >> (4 - w)) & 1;
      const float gr = Rr[w][i * 2 + j], gi = Ri[w][i * 2 + j];
      const float nr = pr * gr - pi * gi;
      pi = pr * gi + pi * gr; pr = nr;
    }
    sUhiR[r * LS + c] = pr; sUhiI[r * LS + c] = pi;
    pr = 1.f; pi = 0.f;
#pragma unroll
    for (int w = 5; w < 10; ++w) {         // wire w (5..9) <-> bit (9-w)
      const int sh = 9 - w;
      const int i = (r >> sh) & 1, j = (c >> sh) & 1;
      const float gr = Rr[w][i * 2 + j], gi = Ri[w][i * 2 + j];
      const float nr = pr * gr - pi * gi;
      pi = pr * gi + pi * gr; pr = nr;
    }
    sUloR[r * LS + c] = pr; sUloI[r * LS + c] = pi;
  }

  // ---- Phase 2: make the staged state row visible as S[32][32] ----
#ifdef USE_ASYNC_LDS
#if __has_builtin(__builtin_amdgcn_s_wait_asynccnt)
  __builtin_amdgcn_s_wait_asynccnt(0);
#else
  asm volatile("s_wait_asynccnt 0" ::: "memory");
#endif
#else
#pragma unroll
  for (int chunk = 0; chunk < 8; ++chunk) {
    const int j = chunk * 128 + lane * 4;
    const float4 v = *(const float4*)(row + j);
    float* d = Xr + (j >> 5) * LS + (j & 31);
    d[0] = v.x; d[1] = v.y; d[2] = v.z; d[3] = v.w;
  }
#endif
  __syncthreads();

  // ---- Phase 3: T = Uhi * S   (S real -> 2 real GEMMs) ----
  cgemm<false, false>(sUhiR, sUhiI, Xr, Xr, Yr, Yi, lm, half);
  __syncthreads();

  // ---- Phase 4: W^T = Ulo * T^T  (=> X[c][r] = post-rotation amplitude a[r*32+c]) ----
  cgemm<true, true>(sUloR, sUloI, Yr, Yi, Xr, Xi, lm, half);
  __syncthreads();

  // ---- Phase 5: CNOT ring folded into the Pauli reduction ----
  // CNOT chain (0,1)..(8,9),(9,0) is linear over GF(2): final wire-0 bit of
  // basis state x is parity(bits 8..0 of x); the wire-0 pair partner of x is
  // x ^ 0x300 (flip bits 9,8 == flip bits 4,3 of r).
  float zr = 0.f, zi = 0.f, ez = 0.f;
  const int r0 = lane;
  const int r1 = lane ^ 24;
  const int pbase = __popc(lane & 15) & 1;
#pragma unroll
  for (int c = 0; c < 32; ++c) {
    const float ar = Xr[c * LS + r0], ai = Xi[c * LS + r0];
    const float m = ar * ar + ai * ai;
    const int p = (pbase + __popc(c)) & 1;
    ez += p ? -m : m;
    if (!p) {  // x is an |0> slot; partner is its |1> slot
      const float br = Xr[c * LS + r1], bi = Xi[c * LS + r1];
      zr += ar * br + ai * bi;   // Re(conj(a0)*a1)
      zi += ar * bi - ai * br;   // Im(conj(a0)*a1)
    }
  }
#pragma unroll
  for (int off = 16; off >= 1; off >>= 1) {
    zr += __shfl_xor(zr, off, 32);
    zi += __shfl_xor(zi, off, 32);
    ez += __shfl_xor(ez, off, 32);
  }
  if (lane == 0 && b < B) {
    out[b * 3 + 0] = fixv(2.f * zr);
    out[b * 3 + 1] = fixv(2.f * zi);
    out[b * 3 + 2] = fixv(ez);
  }
}

extern "C" void kernel_launch(void* const* d_in, const int* in_sizes, int n_in,
                              void* d_out, int out_size, void* d_ws, size_t ws_size,
                              hipStream_t stream) {
  (void)n_in; (void)out_size; (void)d_ws; (void)ws_size;
  const float* patch = (const float*)d_in[0];
  const float* wts   = (const float*)d_in[1];
  float* out = (float*)d_out;
  const int B = in_sizes[0] / 1024;
  const int grid = (B + WPB - 1) / WPB;
  qcirc_kernel<<<grid, WPB * 32, 0, stream>>>(patch, wts, out, B);
}